// USReconstructionLoss_59596966199719
// MI455X (gfx1250) — compile-verified
//
#include <hip/hip_runtime.h>
#include <cstdint>

#define ALPHA_C   0.1f
#define NORM_EPS_C 1e-6f
#define WPB   8           // waves (patches) per block
#define TPB   (WPB * 32)  // 256 threads
#define WMAX_S 512        // static max window size (W_max from the reference)

typedef __attribute__((address_space(1))) int gint_t;  // global (prints as __device__ int*)
typedef __attribute__((address_space(3))) int lint_t;  // LDS

__device__ __forceinline__ float wave_red_add(float v) {
#pragma unroll
  for (int m = 16; m >= 1; m >>= 1) v += __shfl_xor(v, m, 32);
  return v;  // full butterfly: every lane holds the wave sum
}

__global__ void __launch_bounds__(TPB)
us_loss_main(const float* __restrict__ pred,
             const float* __restrict__ signal,
             const int*   __restrict__ mask,
             const int*   __restrict__ wsz,
             const int*   __restrict__ pad,
             const int*   __restrict__ slen,
             float*       __restrict__ part,
             int B, int S, int Wmax, int T)
{
  __shared__ float ssig[WPB][WMAX_S];  // 16 KB: per-wave signal patch
  __shared__ float acc[WPB][4];

  const int tid  = threadIdx.x;
  const int lane = tid & 31;
  const int wave = tid >> 5;
  const int b    = blockIdx.x;
  const int s    = blockIdx.y * WPB + wave;

  float ml = 0.f, vl = 0.f, mc = 0.f, vc = 0.f;

  if (s < S) {
    const int  W    = wsz[b];
    const int  np   = slen[b] / W;
    const bool real = (pad[(size_t)b * S + s] > 0) && (s < np);
    if (real) {
      const int    nIter   = (W + 31) >> 5;          // <= 16
      const size_t sigBase = (size_t)b * T + (size_t)s * W;
      float*       lds     = &ssig[wave][0];

      // ---- stage signal patch into LDS via CDNA5 async-to-LDS path ----
#if __has_builtin(__builtin_amdgcn_global_load_async_to_lds_b32)
#pragma unroll 1
      for (int i = 0; i < nIter; ++i) {
        const int w = lane + (i << 5);
        if (w < W) {
          __builtin_amdgcn_global_load_async_to_lds_b32(
              (gint_t*)(uintptr_t)(signal + sigBase + w),
              (lint_t*)(unsigned int)(uintptr_t)(lds + w),
              0, 0);
        }
      }
#if __has_builtin(__builtin_amdgcn_s_wait_asynccnt)
      __builtin_amdgcn_s_wait_asynccnt(0);
#else
      asm volatile("s_wait_asynccnt 0" ::: "memory");
#endif
      asm volatile("" ::: "memory");
#else
#pragma unroll 1
      for (int i = 0; i < nIter; ++i) {
        const int w = lane + (i << 5);
        if (w < W) lds[w] = signal[sigBase + w];
      }
#endif

      // ---- pass 1: mean (exact) ----
      float tv[16];
      float lsum = 0.f;
#pragma unroll
      for (int i = 0; i < 16; ++i) {
        const int w = lane + (i << 5);
        float t = (w < W) ? lds[w] : 0.f;
        tv[i] = t;
        lsum += t;
      }
      const float mu = wave_red_add(lsum) / (float)W;

      // ---- pass 2: unbiased variance (exact two-pass like the reference) ----
      float lss = 0.f;
#pragma unroll
      for (int i = 0; i < 16; ++i) {
        const int w = lane + (i << 5);
        if (w < W) { const float dt = tv[i] - mu; lss += dt * dt; }
      }
      float var = wave_red_add(lss) / (float)(W - 1);
      var = fmaxf(var, 0.f);
      const float sd  = fmaxf(sqrtf(var), NORM_EPS_C);
      const float inv = 1.f / sd;

      // ---- pass 3: smooth-L1 vs pred (NT loads: pred touched exactly once) ----
      const size_t pBase = ((size_t)b * S + (size_t)s) * (size_t)Wmax;
      float lel = 0.f;
#pragma unroll
      for (int i = 0; i < 16; ++i) {
        const int w = lane + (i << 5);
        if (w < W) {
          const float p  = __builtin_nontemporal_load(pred + pBase + w);
          const float d  = p - (tv[i] - mu) * inv;
          const float ad = fabsf(d);
          lel += (ad < 1.f) ? (0.5f * d * d) : (ad - 0.5f);
        }
      }
      const float pl = wave_red_add(lel) / (float)W;

      const int m = mask[(size_t)b * S + s];
      if (m > 0) { ml = pl; mc = 1.f; }
      else       { vl = pl; vc = 1.f; }
    }
  }

  if (lane == 0) {
    acc[wave][0] = ml; acc[wave][1] = vl; acc[wave][2] = mc; acc[wave][3] = vc;
  }
  __syncthreads();
  if (tid == 0) {  // deterministic fixed-order per-block partial
    float a0 = 0.f, a1 = 0.f, a2 = 0.f, a3 = 0.f;
    for (int i = 0; i < WPB; ++i) {
      a0 += acc[i][0]; a1 += acc[i][1]; a2 += acc[i][2]; a3 += acc[i][3];
    }
    const size_t bid = (size_t)blockIdx.y * gridDim.x + blockIdx.x;
    part[bid * 4 + 0] = a0; part[bid * 4 + 1] = a1;
    part[bid * 4 + 2] = a2; part[bid * 4 + 3] = a3;
  }
}

__global__ void __launch_bounds__(256)
us_loss_final(const float* __restrict__ part, int nblocks, float* __restrict__ out)
{
  __shared__ float r0[256], r1[256], r2[256], r3[256];
  const int t = threadIdx.x;
  float a0 = 0.f, a1 = 0.f, a2 = 0.f, a3 = 0.f;
  for (int i = t; i < nblocks; i += 256) {
    a0 += part[(size_t)i * 4 + 0];
    a1 += part[(size_t)i * 4 + 1];
    a2 += part[(size_t)i * 4 + 2];
    a3 += part[(size_t)i * 4 + 3];
  }
  r0[t] = a0; r1[t] = a1; r2[t] = a2; r3[t] = a3;
  __syncthreads();
  for (int off = 128; off > 0; off >>= 1) {
    if (t < off) {
      r0[t] += r0[t + off]; r1[t] += r1[t + off];
      r2[t] += r2[t + off]; r3[t] += r3[t + off];
    }
    __syncthreads();
  }
  if (t == 0) {
    const float masked  = r0[0] / fmaxf(r2[0], 1.f);
    const float visible = r1[0] / fmaxf(r3[0], 1.f);
    out[0] = masked + ALPHA_C * visible;  // loss
    out[1] = masked;                      // masked_loss
    out[2] = visible;                     // visible_loss
  }
}

extern "C" void kernel_launch(void* const* d_in, const int* in_sizes, int n_in,
                              void* d_out, int out_size, void* d_ws, size_t ws_size,
                              hipStream_t stream) {
  const float* pred   = (const float*)d_in[0];
  const float* signal = (const float*)d_in[1];
  const int*   mask   = (const int*)d_in[2];
  const int*   wsz    = (const int*)d_in[3];
  const int*   pad    = (const int*)d_in[4];
  const int*   slen   = (const int*)d_in[5];

  const int B    = in_sizes[3];
  const int S    = in_sizes[2] / B;
  const int Wmax = in_sizes[0] / in_sizes[2];
  const int T    = in_sizes[1] / B;

  const int SB = (S + WPB - 1) / WPB;
  dim3 grid(B, SB);
  float* part = (float*)d_ws;  // B*SB*4 floats = 128 KB of scratch

  us_loss_main<<<grid, TPB, 0, stream>>>(pred, signal, mask, wsz, pad, slen,
                                         part, B, S, Wmax, T);
  us_loss_final<<<1, 256, 0, stream>>>(part, B * SB, (float*)d_out);
}